// AMFAR_38130719653975
// MI455X (gfx1250) — compile-verified
//
#include <hip/hip_runtime.h>

typedef __attribute__((ext_vector_type(2))) float v2f;
typedef __attribute__((ext_vector_type(8))) float v8f;

static constexpr int Cn = 64;      // prototypes (context rows)
static constexpr int Dn = 512;     // feature dim
static constexpr int Qn = 65536;   // queries (target rows)
static constexpr int KB = 128;     // K-slice staged in LDS (64 KB per slice)
static constexpr int NS = Dn / KB; // 4 slices
static constexpr int SLICE_FLOATS = 2 * Cn * KB;        // 16384 floats = 64 KB
static constexpr int SLICE_CHUNKS = SLICE_FLOATS / 4;   // 4096 x b128

static __device__ __forceinline__ v8f wmma_f32_16x16x4(v2f a, v2f b, v8f c) {
  // D = A(16x4, f32) * B(4x16, f32) + C(16x16, f32)
  return __builtin_amdgcn_wmma_f32_16x16x4_f32(false, a, false, b, (short)0, c,
                                               false, false);
}

// Issue async copies of one K-slice of both contexts into LDS at lds_byte_base.
// LDS layout within a slice: float index = mod*Cn*KB + row*KB + klocal
// (== chunk*16 bytes). 16 async-b128 instructions per wave per slice.
static __device__ __forceinline__ void stage_slice(const float* __restrict__ Pr,
                                                   const float* __restrict__ Pf,
                                                   int kb, unsigned lds_byte_base,
                                                   int tid) {
  for (int c = tid; c < SLICE_CHUNKS; c += 256) {
    const int mod = c >> 11;          // 2048 chunks per modality
    const int r   = (c >> 5) & 63;    // 32 chunks (128 floats) per row
    const int kl  = (c & 31) << 2;
    const float* g = (mod ? Pf : Pr) + (size_t)r * Dn + kb + kl;
    const unsigned lds_off = lds_byte_base + (unsigned)c * 16u;
    asm volatile("global_load_async_to_lds_b128 %0, %1, off"
                 :
                 : "v"(lds_off), "v"(g)
                 : "memory");
  }
}

// ---------------------------------------------------------------------------
// Kernel 0: zero the scalar accumulators, compute ||p||^2 for both contexts.
// ---------------------------------------------------------------------------
__global__ void amfar_init(const float* __restrict__ Pr,
                           const float* __restrict__ Pf,
                           float* __restrict__ scal,     // 8 floats
                           float* __restrict__ pnorm) {  // 128 floats (r then f)
  const int t = threadIdx.x;  // 0..127
  if (t < 8) scal[t] = 0.0f;
  const float* src = (t < 64) ? (Pr + (size_t)t * Dn)
                              : (Pf + (size_t)(t - 64) * Dn);
  float s = 0.0f;
  for (int k = 0; k < Dn; k += 4) {
    float4 v = *(const float4*)(src + k);
    s += v.x * v.x + v.y * v.y + v.z * v.z + v.w * v.w;
  }
  pnorm[t] = s;
}

// ---------------------------------------------------------------------------
// Kernel 1: main fused pass. 8 waves/block; one wave = 16 queries, both
// modalities. Contexts staged K-slice-wise into double-buffered LDS via
// ASYNCcnt-pipelined async-to-LDS copies; queries streamed from HBM as WMMA
// A-fragments.
// ---------------------------------------------------------------------------
__global__ __launch_bounds__(256, 1)
void amfar_main(const float* __restrict__ Pr, const float* __restrict__ Pf,
                const float* __restrict__ Tr, const float* __restrict__ Tf,
                const float* __restrict__ pnorm,
                float* __restrict__ out_pf, float* __restrict__ out_pr,
                float* __restrict__ out_post,
                float* __restrict__ wkl_fr, float* __restrict__ wkl_rf,
                unsigned int* __restrict__ nr_ctr,
                float* __restrict__ sum_cr, float* __restrict__ sum_cf) {
  // Dynamic LDS: 2 buffers x 64 KB = 128 KB. No static LDS in this kernel, so
  // the dynamic segment starts at LDS byte offset 0 (used by the async copies).
  extern __shared__ float smem[];

  const int tid  = threadIdx.x;
  const int lane = tid & 31;
  const int wave = tid >> 5;                 // 0..7
  const int q0   = (blockIdx.x * 8 + wave) * 16;
  const int m    = lane & 15;   // row (A) / col (B,C,D) index within 16
  const int hi   = lane >> 4;   // half-wave select
  const int kh   = hi * 2;      // K sub-pair: lanes 0-15 -> K{0,1}, 16-31 -> K{2,3}

  const float* qrowR = Tr + (size_t)(q0 + m) * Dn + kh;
  const float* qrowF = Tf + (size_t)(q0 + m) * Dn + kh;
  int sBoff[4];
#pragma unroll
  for (int t = 0; t < 4; ++t) sBoff[t] = (t * 16 + m) * KB + kh;

  v8f accR[4] = {}, accF[4] = {};
  float qnR = 0.0f, qnF = 0.0f;

  // Async pipeline prologue: slice 0 -> buffer 0.
  stage_slice(Pr, Pf, 0, 0u, tid);

  for (int s = 0; s < NS; ++s) {
    const int cur = (s & 1) * SLICE_FLOATS;
    if (s + 1 < NS) {
      // Issue next slice into the other buffer (freed by the barrier that
      // ended iteration s-1), then wait until only those 16 copies remain
      // in flight: async loads complete in order, so slice s has landed.
      stage_slice(Pr, Pf, (s + 1) * KB,
                  (unsigned)(((s + 1) & 1) * SLICE_FLOATS * 4), tid);
      asm volatile("s_wait_asynccnt 0x10" ::: "memory");
    } else {
      asm volatile("s_wait_asynccnt 0x0" ::: "memory");
    }
    __syncthreads();  // all waves' copies of slice s visible

    // S += Q_tile (16xKB) * P_slice^T (KBx64), K in steps of 4 via f32 WMMA.
    const int kb = s * KB;
#pragma unroll 4
    for (int k0 = 0; k0 < KB; k0 += 4) {
      v2f aR = *(const v2f*)(qrowR + kb + k0);
      v2f aF = *(const v2f*)(qrowF + kb + k0);
      qnR += aR.x * aR.x + aR.y * aR.y;
      qnF += aF.x * aF.x + aF.y * aF.y;
#pragma unroll
      for (int t = 0; t < 4; ++t) {
        v2f bR = *(const v2f*)&smem[cur + sBoff[t] + k0];
        v2f bF = *(const v2f*)&smem[cur + Cn * KB + sBoff[t] + k0];
        accR[t] = wmma_f32_16x16x4(aR, bR, accR[t]);
        accF[t] = wmma_f32_16x16x4(aF, bF, accF[t]);
      }
    }
    __syncthreads();  // slice s consumed; its buffer may be overwritten
  }

  // ||q||^2: lane m holds K%4 in {kh,kh+1}; partner lane m+16 holds the rest.
  qnR += __shfl_down(qnR, 16, 32);
  qnF += __shfl_down(qnF, 16, 32);
  float qRowR[8], qRowF[8];
#pragma unroll
  for (int j = 0; j < 8; ++j) {
    const int src = hi * 8 + j;  // row of C/D element (VGPR j, this half)
    qRowR[j] = __shfl(qnR, src, 32);
    qRowF[j] = __shfl(qnF, src, 32);
  }
  float pnR[4], pnF[4];
#pragma unroll
  for (int t = 0; t < 4; ++t) {
    pnR[t] = pnorm[t * 16 + m];
    pnF[t] = pnorm[64 + t * 16 + m];
  }

  // Pass 1: d = sqrt(max(||q||^2 - 2s + ||p||^2, 0)), e = exp(-d); row stats.
  float dR[4][8], dF[4][8];
  float Er[8], Ef[8], meR[8], meF[8], sedR[8], sedF[8];
#pragma unroll
  for (int j = 0; j < 8; ++j) {
    Er[j] = 0.f; Ef[j] = 0.f; meR[j] = 0.f; meF[j] = 0.f;
    sedR[j] = 0.f; sedF[j] = 0.f;
  }
#pragma unroll
  for (int t = 0; t < 4; ++t) {
#pragma unroll
    for (int j = 0; j < 8; ++j) {
      float d2r = qRowR[j] + pnR[t] - 2.0f * accR[t][j];
      float dr  = sqrtf(fmaxf(d2r, 0.0f));
      float er  = __expf(-dr);
      accR[t][j] = er; dR[t][j] = dr;
      Er[j] += er; meR[j] = fmaxf(meR[j], er); sedR[j] += er * dr;

      float d2f = qRowF[j] + pnF[t] - 2.0f * accF[t][j];
      float df  = sqrtf(fmaxf(d2f, 0.0f));
      float ef  = __expf(-df);
      accF[t][j] = ef; dF[t][j] = df;
      Ef[j] += ef; meF[j] = fmaxf(meF[j], ef); sedF[j] += ef * df;
    }
  }

  // Reduce across the 16 lanes of each half-wave (rows live per half).
#pragma unroll
  for (int j = 0; j < 8; ++j) {
#pragma unroll
    for (int off = 1; off < 16; off <<= 1) {
      Er[j]   += __shfl_xor(Er[j],   off, 32);
      Ef[j]   += __shfl_xor(Ef[j],   off, 32);
      sedR[j] += __shfl_xor(sedR[j], off, 32);
      sedF[j] += __shfl_xor(sedF[j], off, 32);
      meR[j]   = fmaxf(meR[j], __shfl_xor(meR[j], off, 32));
      meF[j]   = fmaxf(meF[j], __shfl_xor(meF[j], off, 32));
    }
  }

  // Per-row quantities: log-sum, confidence, entropy, fusion weights.
  float invEr[8], invEf[8], lEr[8], lEf[8];
  float cR[8], cF[8], hR[8], hF[8], wr[8], wf[8], invPs[8];
#pragma unroll
  for (int j = 0; j < 8; ++j) {
    invEr[j] = 1.0f / Er[j]; lEr[j] = __logf(Er[j]);
    invEf[j] = 1.0f / Ef[j]; lEf[j] = __logf(Ef[j]);
    cR[j] = meR[j] * invEr[j];
    cF[j] = meF[j] * invEf[j];
    hR[j] = sedR[j] * invEr[j] + lEr[j];  // -sum p log p
    hF[j] = sedF[j] * invEf[j] + lEf[j];
    const float s = cR[j] + cF[j];
    wr[j] = cR[j] / s;
    wf[j] = cF[j] / s;
    invPs[j] = 1.0f / (wr[j] * Er[j] + wf[j] * Ef[j]);
  }

  // Pass 2: posteriors, fused posterior, "KL" terms (faithful: log(p)-p form).
  float klfr[8], klrf[8];
#pragma unroll
  for (int j = 0; j < 8; ++j) { klfr[j] = 0.f; klrf[j] = 0.f; }
#pragma unroll
  for (int t = 0; t < 4; ++t) {
#pragma unroll
    for (int j = 0; j < 8; ++j) {
      const float er = accR[t][j], ef = accF[t][j];
      const float pr = er * invEr[j];
      const float pf = ef * invEf[j];
      klfr[j] += pf * ((-dF[t][j] - lEf[j]) - pr);  // p_f*(log p_f - p_r)
      klrf[j] += pr * ((-dR[t][j] - lEr[j]) - pf);  // p_r*(log p_r - p_f)
      const float fused = (wr[j] * er + wf[j] * ef) * invPs[j];
      const size_t idx = (size_t)(q0 + hi * 8 + j) * Cn + (t * 16 + m);
      out_pf[idx]   = pf;
      out_pr[idx]   = pr;
      out_post[idx] = fused;
    }
  }
#pragma unroll
  for (int j = 0; j < 8; ++j) {
#pragma unroll
    for (int off = 1; off < 16; off <<= 1) {
      klfr[j] += __shfl_xor(klfr[j], off, 32);
      klrf[j] += __shfl_xor(klrf[j], off, 32);
    }
  }

  // One writer lane per half-wave handles its 8 rows' scalar side-outputs.
  if (m == 0) {
    unsigned cnt = 0;
    float lcr = 0.0f, lcf = 0.0f;
#pragma unroll
    for (int j = 0; j < 8; ++j) {
      const int row = q0 + hi * 8 + j;
      wkl_fr[row] = cF[j] * (klfr[j] * (1.0f / 64.0f));
      wkl_rf[row] = cR[j] * (klrf[j] * (1.0f / 64.0f));
      cnt += ((hR[j] > hF[j]) && (cR[j] > cF[j])) ? 1u : 0u;
      lcr += cR[j];
      lcf += cF[j];
    }
    atomicAdd(nr_ctr, cnt);
    atomicAdd(sum_cr, lcr);
    atomicAdd(sum_cf, lcf);
  }
}

// ---------------------------------------------------------------------------
// Kernel 2: prefix-masked sums (need global n_r first, hence separate kernel).
// ---------------------------------------------------------------------------
__global__ __launch_bounds__(256)
void amfar_reduce(const float* __restrict__ wkl_fr,
                  const float* __restrict__ wkl_rf,
                  const unsigned int* __restrict__ nr_ctr,
                  float* __restrict__ accLfr, float* __restrict__ accLrf) {
  const int i = blockIdx.x * blockDim.x + threadIdx.x;
  const unsigned nr = *nr_ctr;
  const unsigned nf = (unsigned)Qn - nr;
  float vf = ((unsigned)i < nf) ? wkl_fr[i] : 0.0f;
  float vr = ((unsigned)i < nr) ? wkl_rf[i] : 0.0f;
#pragma unroll
  for (int off = 16; off > 0; off >>= 1) {
    vf += __shfl_down(vf, off, 32);
    vr += __shfl_down(vr, off, 32);
  }
  if ((threadIdx.x & 31) == 0) {
    atomicAdd(accLfr, vf);
    atomicAdd(accLrf, vr);
  }
}

// ---------------------------------------------------------------------------
// Kernel 3: final scalars.
// ---------------------------------------------------------------------------
__global__ void amfar_final(const float* __restrict__ scal,
                            float* __restrict__ out) {
  if (threadIdx.x == 0) {
    out[0] = scal[3] / scal[2];  // L_f_r = acc / sum(c_f)
    out[1] = scal[4] / scal[1];  // L_r_f = acc / sum(c_r)
  }
}

// ---------------------------------------------------------------------------
extern "C" void kernel_launch(void* const* d_in, const int* in_sizes, int n_in,
                              void* d_out, int out_size, void* d_ws,
                              size_t ws_size, hipStream_t stream) {
  (void)in_sizes; (void)n_in; (void)out_size; (void)ws_size;
  const float* Pr = (const float*)d_in[0];  // context_rgb  [64,512]
  const float* Pf = (const float*)d_in[1];  // context_flow [64,512]
  const float* Tr = (const float*)d_in[2];  // target_rgb   [65536,512]
  const float* Tf = (const float*)d_in[3];  // target_flow  [65536,512]
  float* out = (float*)d_out;
  float* ws  = (float*)d_ws;

  // ws layout (floats):
  //   [0]=n_r (u32 bits) [1]=sum_cr [2]=sum_cf [3]=accLfr [4]=accLrf
  //   [8..136)  pnorm (rgb 64, flow 64)
  //   [256..256+Q)        wkl_fr
  //   [256+Q..256+2Q)     wkl_rf
  float* scal  = ws;
  float* pnorm = ws + 8;
  float* wklfr = ws + 256;
  float* wklrf = ws + 256 + Qn;

  // d_out layout: [0]=L_f_r [1]=L_r_f, then p_f, p_r, post (each Q*C).
  float* out_pf   = out + 2;
  float* out_pr   = out + 2 + (size_t)Qn * Cn;
  float* out_post = out + 2 + 2 * (size_t)Qn * Cn;

  const size_t lds_bytes = 2u * SLICE_FLOATS * sizeof(float);  // 128 KB

  amfar_init<<<1, 128, 0, stream>>>(Pr, Pf, scal, pnorm);
  amfar_main<<<Qn / 128, 256, lds_bytes, stream>>>(Pr, Pf, Tr, Tf, pnorm,
                                                   out_pf, out_pr, out_post,
                                                   wklfr, wklrf,
                                                   (unsigned int*)scal,
                                                   scal + 1, scal + 2);
  amfar_reduce<<<Qn / 256, 256, 0, stream>>>(wklfr, wklrf,
                                             (const unsigned int*)scal,
                                             scal + 3, scal + 4);
  amfar_final<<<1, 1, 0, stream>>>(scal, out);
}